// HamiltonianBlockGenLayer_14044543058109
// MI455X (gfx1250) — compile-verified
//
#include <hip/hip_runtime.h>
#include <hip/hip_bf16.h>

// ---------------- problem constants (match reference) ----------------
#define NAT   1000          // atoms
#define BB_   14            // orbitals per atom
#define FF    16            // node feature size
#define HIDN  64            // MLP hidden
#define PP    32000         // pairs
#define NB    14000         // NAT * BB_
#define BLK   196           // BB_*BB_
#define TPAIR 10

typedef float v2f __attribute__((ext_vector_type(2)));
typedef float v4f __attribute__((ext_vector_type(4)));
typedef float v8f __attribute__((ext_vector_type(8)));

// ---------------- workspace layout (ints) ----------------
#define OFF_CNT    0
#define OFF_BASE   16
#define OFF_CUR    32
#define OFF_NTILES 48
#define OFF_TTYPE  64
#define OFF_TSTART (64 + 2048)
#define OFF_TCNT   (64 + 4096)
#define OFF_ORDER  (64 + 6144)
#define MAX_TILES  2016

// ---------------- bulk copy H_init -> out (bandwidth bound, ~1.57 GB) ----------------
__global__ void k_copy(const v4f* __restrict__ src, v4f* __restrict__ dst, long n4) {
  long i = (long)blockIdx.x * blockDim.x + threadIdx.x;
  long stride = (long)gridDim.x * blockDim.x;
  for (; i < n4; i += stride) {
    v4f v = __builtin_nontemporal_load(src + i);
    __builtin_nontemporal_store(v, dst + i);
  }
}

// ---------------- pair-type bucketing ----------------
__global__ void k_ws_zero(int* ws) {
  int i = threadIdx.x;
  if (i < 64) ws[i] = 0;
}

__global__ void k_hist(const int* __restrict__ pair_type, int* ws) {
  int p = blockIdx.x * blockDim.x + threadIdx.x;
  if (p < PP) atomicAdd(&ws[OFF_CNT + pair_type[p]], 1);
}

__global__ void k_build(int* ws) {
  if (threadIdx.x != 0 || blockIdx.x != 0) return;
  int total = 0, nt = 0;
  for (int t = 0; t < TPAIR; ++t) {
    ws[OFF_BASE + t] = total;
    int c = ws[OFF_CNT + t];
    int tiles = (c + 15) >> 4;
    for (int k = 0; k < tiles; ++k) {
      ws[OFF_TTYPE + nt]  = t;
      ws[OFF_TSTART + nt] = total + k * 16;
      int rem = c - k * 16;
      ws[OFF_TCNT + nt]   = rem < 16 ? rem : 16;
      ++nt;
    }
    total += c;
  }
  ws[OFF_NTILES] = nt;
}

__global__ void k_scatter(const int* __restrict__ pair_type, int* ws) {
  int p = blockIdx.x * blockDim.x + threadIdx.x;
  if (p < PP) {
    int t = pair_type[p];
    int pos = atomicAdd(&ws[OFF_CUR + t], 1);
    ws[OFF_ORDER + ws[OFF_BASE + t] + pos] = p;
  }
}

// ---------------- off-diagonal MLP: 16 same-type pairs per wave via WMMA f32 ----------------
__global__ __launch_bounds__(32) void k_pair_wmma(
    const float* __restrict__ nf, const float* __restrict__ cmask,
    const float* __restrict__ off_blk,
    const float* __restrict__ Wo1, const float* __restrict__ bo1,
    const float* __restrict__ Wo2, const float* __restrict__ bo2,
    const int* __restrict__ pair_i, const int* __restrict__ pair_j,
    const int* __restrict__ ws, float* __restrict__ out)
{
  __shared__ float hp[16][37];   // input tile, 36 cols used (33 real + pad), stride 37
  __shared__ float zs[16][65];   // hidden tile, 64 cols used, stride 65
  __shared__ float w1t[4][64];   // W1 K-tail rows 32..35 (rows 33..35 zero)
  __shared__ float w2t[64][16];  // W2 N-tail cols 192..207 (cols 196..207 zero)

  int ntiles = ws[OFF_NTILES];
  int bid = blockIdx.x;
  if (bid >= ntiles) return;
  int t     = ws[OFF_TTYPE + bid];
  int start = ws[OFF_TSTART + bid];
  int cnt   = ws[OFF_TCNT + bid];
  const int* order = ws + OFF_ORDER;

  const float* W1 = Wo1 + t * (33 * HIDN);
  const float* W2 = Wo2 + t * (HIDN * BLK);

  int lane = threadIdx.x;

  // cooperative gather of input tile hp[16 pairs][36]: [nf_i | nf_j | e | 0 0 0]
  for (int idx = lane; idx < 16 * 36; idx += 32) {
    int row = idx / 36, col = idx - row * 36;
    float v = 0.f;
    if (row < cnt && col < 33) {
      int p = order[start + row];
      int i = pair_i[p], j = pair_j[p];
      if (col < 16)      v = nf[i * FF + col];
      else if (col < 32) v = nf[j * FF + (col - 16)];
      else               v = cmask[(long)i * NAT + j];
    }
    hp[row][col] = v;
  }
  // stage zero-padded W1 K-tail (rows 32..35)
  for (int idx = lane; idx < 4 * 64; idx += 32) {
    int r = idx >> 6, c = idx & 63;
    w1t[r][c] = (r == 0) ? W1[32 * HIDN + c] : 0.f;
  }
  // stage zero-padded W2 N-tail (cols 192..207)
  for (int idx = lane; idx < 64 * 16; idx += 32) {
    int r = idx >> 4, c = idx & 15;
    w2t[r][c] = (c < 4) ? W2[r * BLK + 192 + c] : 0.f;
  }
  __syncthreads();

  int half = lane >> 4;      // A/B: lanes 0-15 carry K=+0/+1, lanes 16-31 carry K=+2/+3
  int l16  = lane & 15;

  // ---- layer 1: [16x36] x [36x64] ----
  v8f acc[4] = {};
#pragma unroll
  for (int kk = 0; kk < 8; ++kk) {              // K = 0..31, guard-free
    int k0 = kk * 4 + half * 2;
    v2f a;
    a.x = hp[l16][k0];
    a.y = hp[l16][k0 + 1];
#pragma unroll
    for (int nn = 0; nn < 4; ++nn) {
      int colw = nn * 16 + l16;
      v2f b;
      b.x = W1[k0 * HIDN + colw];
      b.y = W1[(k0 + 1) * HIDN + colw];
      acc[nn] = __builtin_amdgcn_wmma_f32_16x16x4_f32(
          false, a, false, b, (short)0, acc[nn], false, false);
    }
  }
  {                                             // K-tail 32..35 from padded LDS
    int k0 = 32 + half * 2;
    v2f a;
    a.x = hp[l16][k0];
    a.y = hp[l16][k0 + 1];
#pragma unroll
    for (int nn = 0; nn < 4; ++nn) {
      int colw = nn * 16 + l16;
      v2f b;
      b.x = w1t[half * 2][colw];
      b.y = w1t[half * 2 + 1][colw];
      acc[nn] = __builtin_amdgcn_wmma_f32_16x16x4_f32(
          false, a, false, b, (short)0, acc[nn], false, false);
    }
  }

  // bias + relu, stage z to LDS (C/D layout: VGPR r -> M = r + 8*half, lane%16 -> N)
#pragma unroll
  for (int nn = 0; nn < 4; ++nn) {
    int colz = nn * 16 + l16;
    float bias = bo1[t * HIDN + colz];
#pragma unroll
    for (int r = 0; r < 8; ++r) {
      int M = r + half * 8;
      float v = acc[nn][r] + bias;
      zs[M][colz] = v > 0.f ? v : 0.f;
    }
  }
  __syncthreads();

  // preload per-row bookkeeping once (instead of 13x in the n-tile loop)
  int row0 = half * 8;
  bool vrow[8]; int irow[8], jrow[8]; long pofs[8];
#pragma unroll
  for (int r = 0; r < 8; ++r) {
    int M = r + row0;
    vrow[r] = (M < cnt);
    int p = vrow[r] ? order[start + M] : 0;
    irow[r] = pair_i[p];
    jrow[r] = pair_j[p];
    pofs[r] = (long)p * BLK;
  }

  // ---- layer 2: [16x64] x [64x196] ----
  for (int nn2 = 0; nn2 < 12; ++nn2) {          // full n-tiles, guard-free
    int colw = nn2 * 16 + l16;
    v8f acc2 = {};
#pragma unroll
    for (int kk = 0; kk < 16; ++kk) {
      int k0 = kk * 4 + half * 2;
      v2f a;
      a.x = zs[l16][k0];
      a.y = zs[l16][k0 + 1];
      v2f b;
      b.x = W2[k0 * BLK + colw];
      b.y = W2[(k0 + 1) * BLK + colw];
      acc2 = __builtin_amdgcn_wmma_f32_16x16x4_f32(
          false, a, false, b, (short)0, acc2, false, false);
    }
    int rb = colw / BB_, cb = colw - rb * BB_;
    float bias = bo2[t * BLK + colw];
#pragma unroll
    for (int r = 0; r < 8; ++r) {
      if (vrow[r]) {
        float v = (acc2[r] + bias) * off_blk[pofs[r] + colw];
        out[((long)irow[r] * BB_ + rb) * NB + (long)jrow[r] * BB_ + cb] = v;
        out[((long)jrow[r] * BB_ + cb) * NB + (long)irow[r] * BB_ + rb] = v;
      }
    }
  }
  {                                             // N-tail (cols 192..195) from padded LDS
    v8f acc2 = {};
#pragma unroll
    for (int kk = 0; kk < 16; ++kk) {
      int k0 = kk * 4 + half * 2;
      v2f a;
      a.x = zs[l16][k0];
      a.y = zs[l16][k0 + 1];
      v2f b;
      b.x = w2t[k0][l16];
      b.y = w2t[k0 + 1][l16];
      acc2 = __builtin_amdgcn_wmma_f32_16x16x4_f32(
          false, a, false, b, (short)0, acc2, false, false);
    }
    int colw = 192 + l16;
    if (colw < BLK) {
      int rb = colw / BB_, cb = colw - rb * BB_;
      float bias = bo2[t * BLK + colw];
#pragma unroll
      for (int r = 0; r < 8; ++r) {
        if (vrow[r]) {
          float v = (acc2[r] + bias) * off_blk[pofs[r] + colw];
          out[((long)irow[r] * BB_ + rb) * NB + (long)jrow[r] * BB_ + cb] = v;
          out[((long)jrow[r] * BB_ + cb) * NB + (long)irow[r] * BB_ + rb] = v;
        }
      }
    }
  }
}

// ---------------- diagonal MLP (27 MFLOP total, plain VALU) ----------------
__global__ void k_atom(const float* __restrict__ nf, const float* __restrict__ cmask,
                       const float* __restrict__ atom_blk,
                       const float* __restrict__ Wd1, const float* __restrict__ bd1,
                       const float* __restrict__ Wd2, const float* __restrict__ bd2,
                       const int* __restrict__ atom_type, float* __restrict__ out)
{
  int n = blockIdx.x * blockDim.x + threadIdx.x;
  if (n >= NAT) return;
  int t = atom_type[n];
  float h[17];
#pragma unroll
  for (int f = 0; f < FF; ++f) h[f] = nf[n * FF + f];
  h[16] = cmask[(long)n * NAT + n];

  const float* W1 = Wd1 + t * (17 * HIDN);
  float z[HIDN];
  for (int k = 0; k < HIDN; ++k) {
    float s = bd1[t * HIDN + k];
#pragma unroll
    for (int f = 0; f < 17; ++f) s += h[f] * W1[f * HIDN + k];
    z[k] = s > 0.f ? s : 0.f;
  }
  const float* W2 = Wd2 + t * (HIDN * BLK);
  const float* ab = atom_blk + (long)n * BLK;
  for (int o = 0; o < BLK; ++o) {
    float s = bd2[t * BLK + o];
    for (int k = 0; k < HIDN; ++k) s += z[k] * W2[k * BLK + o];
    float v = s * ab[o];
    int rb = o / BB_, cb = o - rb * BB_;
    out[((long)n * BB_ + rb) * NB + (long)n * BB_ + cb] = v;
  }
}

// ---------------- launch ----------------
extern "C" void kernel_launch(void* const* d_in, const int* in_sizes, int n_in,
                              void* d_out, int out_size, void* d_ws, size_t ws_size,
                              hipStream_t stream) {
  const float* nf       = (const float*)d_in[0];
  const float* cmask    = (const float*)d_in[1];
  const float* Hinit    = (const float*)d_in[2];
  const float* atom_blk = (const float*)d_in[3];
  const float* off_blk  = (const float*)d_in[4];
  const float* Wd1      = (const float*)d_in[5];
  const float* bd1      = (const float*)d_in[6];
  const float* Wd2      = (const float*)d_in[7];
  const float* bd2      = (const float*)d_in[8];
  const float* Wo1      = (const float*)d_in[9];
  const float* bo1      = (const float*)d_in[10];
  const float* Wo2      = (const float*)d_in[11];
  const float* bo2      = (const float*)d_in[12];
  const int* atom_type  = (const int*)d_in[13];
  const int* pair_type  = (const int*)d_in[14];
  const int* pair_i     = (const int*)d_in[15];
  const int* pair_j     = (const int*)d_in[16];
  float* out = (float*)d_out;
  int* ws = (int*)d_ws;

  // 1) dominant cost: stream H_init -> out (784 MB each way), float4 non-temporal
  long n4 = (long)NB * NB / 4;
  k_copy<<<8192, 256, 0, stream>>>((const v4f*)Hinit, (v4f*)out, n4);

  // 2) bucket pairs by expert type
  k_ws_zero<<<1, 64, 0, stream>>>(ws);
  k_hist<<<(PP + 255) / 256, 256, 0, stream>>>(pair_type, ws);
  k_build<<<1, 1, 0, stream>>>(ws);
  k_scatter<<<(PP + 255) / 256, 256, 0, stream>>>(pair_type, ws);

  // 3) off-diagonal expert MLPs on WMMA, scatter blocks (after copy by stream order)
  k_pair_wmma<<<MAX_TILES, 32, 0, stream>>>(nf, cmask, off_blk, Wo1, bo1, Wo2, bo2,
                                            pair_i, pair_j, ws, out);

  // 4) diagonal blocks
  k_atom<<<(NAT + 63) / 64, 64, 0, stream>>>(nf, cmask, atom_blk, Wd1, bd1, Wd2, bd2,
                                             atom_type, out);
}